// GATrPairBiasBlock_60146722013684
// MI455X (gfx1250) — compile-verified
//
#include <hip/hip_runtime.h>
#include <math.h>

typedef __attribute__((ext_vector_type(2))) float v2f;
typedef __attribute__((ext_vector_type(4))) float v4f;
typedef __attribute__((ext_vector_type(8))) float v8f;

#define NB 2
#define NN 512
#define NH 8
#define HIDMV 8
#define HIDS 16
#define NSC 64
#define NZ 128
#define NF 144          // 8*16 mv features + 16 scalar features per head
#define LDW 520         // padded logits row stride (even -> 8B aligned pairs)

__constant__ int   GRADE_c[16] = {0,1,1,1,1,2,2,2,2,2,2,3,3,3,3,4};
__constant__ float INNER_c[16] = {1,0,1,1,1,0,0,0,1,1,1,0,0,0,1,0};

__device__ __forceinline__ float gelu_tanh(float x) {
    float x3 = x * x * x;
    return 0.5f * x * (1.0f + tanhf(0.79788456080286535588f * (x + 0.044715f * x3)));
}

// ---------------------------------------------------------------------------
// Kernel A: per-item equi_norm + equi_linear(qkv), split heads, rope(q_s,k_s),
// write Q/K feature-major pair-interleaved [bh][f/2][n][2], V row-major [bh][n][144].
// Scale and INNER folded into Q.
// ---------------------------------------------------------------------------
__global__ __launch_bounds__(384)
void ka_qkv(const float* __restrict__ mv, const float* __restrict__ sc,
            const float* __restrict__ w_mv_qkv, const float* __restrict__ w_s2mv_qkv,
            const float* __restrict__ w_mv2s_qkv, const float* __restrict__ w_s_qkv,
            const float* __restrict__ b_s_qkv,
            float* __restrict__ Qp, float* __restrict__ Kp, float* __restrict__ Vc)
{
    __shared__ float nmv[256];
    __shared__ float ns[64];
    __shared__ float ys[384];
    __shared__ float red[20];

    const int item = blockIdx.x;          // b*NN + n
    const int n = item % NN;
    const int b = item / NN;
    const int tid = threadIdx.x;
    const float* x  = mv + (size_t)item * 256;
    const float* xs = sc + (size_t)item * 64;

    if (tid < 16) {
        float a = 0.f;
        for (int j = 0; j < 16; ++j) { float v = x[tid*16 + j]; a += v * v * INNER_c[j]; }
        red[tid] = a;
    }
    __syncthreads();
    if (tid == 0) {
        float m = 0.f;
        for (int i = 0; i < 16; ++i) m += red[i];
        red[16] = sqrtf(m * (1.0f/16.0f) + 0.01f);
    }
    if (tid == 1) {
        float mu = 0.f;
        for (int i = 0; i < 64; ++i) mu += xs[i];
        mu *= (1.0f/64.0f);
        float v = 0.f;
        for (int i = 0; i < 64; ++i) { float d = xs[i] - mu; v += d * d; }
        v *= (1.0f/64.0f);
        red[17] = mu; red[18] = rsqrtf(v + 1e-5f);
    }
    __syncthreads();
    {
        float invd = 1.0f / red[16];
        for (int idx = tid; idx < 256; idx += 384) nmv[idx] = x[idx] * invd;
        if (tid < 64) ns[tid] = (xs[tid] - red[17]) * red[18];
    }
    __syncthreads();

    const float scale = rsqrtf((float)(8 * HIDMV + HIDS));  // 80^-0.5

    // multivector outputs (192 channels: 3 x 8 heads x 8 hid)
    if (tid < 192) {
        const int o = tid;
        const int t = o / 64, rem = o % 64, h = rem / 8, c = rem % 8;
        const int bh = b * NH + h;
        float y[16];
        #pragma unroll
        for (int xg = 0; xg < 16; ++xg) {
            const float* wr = w_mv_qkv + ((size_t)GRADE_c[xg] * 192 + o) * 16;
            float a = 0.f;
            #pragma unroll
            for (int i = 0; i < 16; ++i) a += wr[i] * nmv[i*16 + xg];
            y[xg] = a;
        }
        {
            const float* wr = w_s2mv_qkv + (size_t)o * 64;
            float a = 0.f;
            for (int s = 0; s < 64; ++s) a += wr[s] * ns[s];
            y[0] += a;
        }
        #pragma unroll
        for (int xg = 0; xg < 16; ++xg) {
            const int f = c * 16 + xg;
            if (t == 0)
                Qp[(((size_t)bh*72 + (f>>1))*NN + n)*2 + (f&1)] = y[xg] * INNER_c[xg] * scale;
            else if (t == 1)
                Kp[(((size_t)bh*72 + (f>>1))*NN + n)*2 + (f&1)] = y[xg];
            else
                Vc[((size_t)bh*NN + n)*NF + f] = y[xg];
        }
    }
    // scalar outputs (384 = 3 x 8 heads x 16)
    {
        const int o = tid;
        float a = b_s_qkv[o];
        const float* wr = w_s_qkv + (size_t)o * 64;
        for (int s = 0; s < 64; ++s) a += wr[s] * ns[s];
        const float* wm = w_mv2s_qkv + (size_t)o * 16;
        #pragma unroll
        for (int i = 0; i < 16; ++i) a += wm[i] * nmv[i*16];
        ys[o] = a;
    }
    __syncthreads();
    {
        const int o = tid;
        const int t = o / 128, rem = o % 128, h = rem / 16, d = rem % 16;
        const int bh = b * NH + h;
        const int base = o - d;
        float val;
        if (t < 2) {
            const int j = (d < 8) ? d : (d - 8);
            float freq = __powf(4096.0f, -(float)j * (1.0f/8.0f));
            float ang = (float)n * freq;
            float cs = __cosf(ang), sn = __sinf(ang);
            if (d < 8) val = ys[base + d] * cs - ys[base + d + 8] * sn;
            else       val = ys[base + j] * sn + ys[base + d] * cs;
        } else {
            val = ys[o];
        }
        const int f = 128 + d;
        if (t == 0)      Qp[(((size_t)bh*72 + (f>>1))*NN + n)*2 + (f&1)] = val * scale;
        else if (t == 1) Kp[(((size_t)bh*72 + (f>>1))*NN + n)*2 + (f&1)] = val;
        else             Vc[((size_t)bh*NN + n)*NF + f] = val;
    }
}

// ---------------------------------------------------------------------------
// Kernel B: pair-scalar LN + per-head projection -> bias[b][h][q][k].
// Pure bandwidth pass over the 256MB pair_scalars tensor: 1 wave per (q,k),
// lane holds 4 z-values (coalesced v4 load), wave32 shfl_xor reductions.
// ---------------------------------------------------------------------------
__global__ __launch_bounds__(256)
void kb_bias(const float* __restrict__ z, const float* __restrict__ g,
             const float* __restrict__ be, const float* __restrict__ w_pb,
             float* __restrict__ bias)
{
    __shared__ float sg[128], sb[128], sw[8*128];
    const int tid = threadIdx.x;
    for (int i = tid; i < 128;  i += 256) { sg[i] = g[i]; sb[i] = be[i]; }
    for (int i = tid; i < 1024; i += 256) sw[i] = w_pb[i];
    __syncthreads();

    const int wave = tid >> 5, lane = tid & 31;
    const long pair = (long)blockIdx.x * 8 + wave;        // < NB*NN*NN
    const int b = (int)(pair / ((long)NN * NN));
    const int rem = (int)(pair % ((long)NN * NN));
    const int q = rem / NN, k = rem % NN;

    v4f v = *(const v4f*)(z + pair * NZ + (size_t)lane * 4);
    float s  = v.x + v.y + v.z + v.w;
    float ss = v.x*v.x + v.y*v.y + v.z*v.z + v.w*v.w;
    for (int m = 16; m >= 1; m >>= 1) { s += __shfl_xor(s, m, 32); ss += __shfl_xor(ss, m, 32); }
    const float mu  = s  * (1.0f/128.0f);
    const float var = ss * (1.0f/128.0f) - mu * mu;
    const float rs  = rsqrtf(var + 1e-5f);

    float zn[4];
    #pragma unroll
    for (int j = 0; j < 4; ++j) zn[j] = (v[j] - mu) * rs * sg[lane*4 + j] + sb[lane*4 + j];

    #pragma unroll
    for (int h = 0; h < 8; ++h) {
        float p = 0.f;
        #pragma unroll
        for (int j = 0; j < 4; ++j) p += zn[j] * sw[h*128 + lane*4 + j];
        for (int m = 16; m >= 1; m >>= 1) p += __shfl_xor(p, m, 32);
        if (lane == 0) bias[(((size_t)b*NH + h)*NN + q)*NN + k] = p;
    }
}

// ---------------------------------------------------------------------------
// Kernel C: attention, one wave per (b,h,q-tile). fp32 WMMA 16x16x4:
//   phase 1: logits tile  = Qcat[16x144] @ Kcat^T  (36 chained wmma / k-tile)
//   phase 2: row softmax over 512 keys in LDS
//   phase 3: O tile chunk = P[16x512] @ Vcat       (128 chained wmma / chunk)
// ---------------------------------------------------------------------------
__global__ __launch_bounds__(32)
void kc_attn(const float* __restrict__ Qp, const float* __restrict__ Kp,
             const float* __restrict__ Vc, const float* __restrict__ bias,
             float* __restrict__ Oc)
{
    __shared__ __align__(16) float lt[16 * LDW];

    const int qt = blockIdx.x, h = blockIdx.y, b = blockIdx.z;
    const int bh = b * NH + h;
    const int q0 = qt * 16;
    const int lane = threadIdx.x;
    const int row = lane & 15;
    const int hi  = lane >> 4;       // 0 or 1
    const int c0  = hi << 1;         // 0 or 2 (fragment K offset)

    const float* Qb = Qp + (size_t)bh * 72 * NN * 2;
    const float* Kb = Kp + (size_t)bh * 72 * NN * 2;
    const float* Bb = bias + (size_t)bh * NN * NN;

    // ---- phase 1: logits ----
    for (int kt = 0; kt < 32; ++kt) {
        const int k0 = kt * 16;
        v8f acc = {};
        #pragma unroll 4
        for (int kk = 0; kk < NF; kk += 4) {
            const int fp = (kk + c0) >> 1;
            v2f a  = *(const v2f*)(Qb + ((size_t)fp * NN + q0 + row) * 2);
            v2f bf = *(const v2f*)(Kb + ((size_t)fp * NN + k0 + row) * 2);
            acc = __builtin_amdgcn_wmma_f32_16x16x4_f32(false, a, false, bf,
                                                        (short)0, acc, false, false);
        }
        if (kt + 1 < 32)
            __builtin_prefetch(Kb + ((size_t)(c0 >> 1) * NN + k0 + 16 + row) * 2, 0, 0);
        #pragma unroll
        for (int r = 0; r < 8; ++r) {
            const int qq = r + hi * 8;
            lt[qq * LDW + k0 + row] = acc[r] + Bb[(size_t)(q0 + qq) * NN + k0 + row];
        }
    }
    __syncthreads();

    // ---- phase 2: softmax (2 lanes per row, 256 cols each) ----
    {
        const int cbase = hi * 256;
        float m = -3.4e38f;
        for (int c = 0; c < 256; ++c) m = fmaxf(m, lt[row * LDW + cbase + c]);
        m = fmaxf(m, __shfl_xor(m, 16, 32));
        float ssum = 0.f;
        for (int c = 0; c < 256; ++c) {
            float e = __expf(lt[row * LDW + cbase + c] - m);
            lt[row * LDW + cbase + c] = e;
            ssum += e;
        }
        ssum += __shfl_xor(ssum, 16, 32);
        const float inv = 1.0f / ssum;
        for (int c = 0; c < 256; ++c) lt[row * LDW + cbase + c] *= inv;
    }
    __syncthreads();

    // ---- phase 3: attn @ Vcat ----
    const float* Vb = Vc + (size_t)bh * NN * NF;
    for (int oc = 0; oc < 9; ++oc) {
        const int f0 = oc * 16;
        v8f acc = {};
        #pragma unroll 4
        for (int k = 0; k < NN; k += 4) {
            const int kb = k + c0;
            v2f a = *(const v2f*)(lt + row * LDW + kb);
            v2f bf;
            bf.x = Vb[(size_t)(kb + 0) * NF + f0 + row];
            bf.y = Vb[(size_t)(kb + 1) * NF + f0 + row];
            acc = __builtin_amdgcn_wmma_f32_16x16x4_f32(false, a, false, bf,
                                                        (short)0, acc, false, false);
        }
        #pragma unroll
        for (int r = 0; r < 8; ++r) {
            const int qq = q0 + r + hi * 8;
            Oc[(((size_t)b * NN + qq) * NH + h) * NF + f0 + row] = acc[r];
        }
    }
}

// ---------------------------------------------------------------------------
// Kernel D: per-item output equi_linear + residual, equi_norm, gated MLP, residual.
// ---------------------------------------------------------------------------
__global__ __launch_bounds__(128)
void kd_post(const float* __restrict__ Oc,
             const float* __restrict__ mv, const float* __restrict__ sc,
             const float* __restrict__ w_mv_out, const float* __restrict__ w_s2mv_out,
             const float* __restrict__ w_mv2s_out, const float* __restrict__ w_s_out,
             const float* __restrict__ b_s_out,
             const float* __restrict__ w_mv1, const float* __restrict__ w_s2mv1,
             const float* __restrict__ w_mv2s1, const float* __restrict__ w_s1,
             const float* __restrict__ b_s1,
             const float* __restrict__ w_mv2, const float* __restrict__ w_s2mv2,
             const float* __restrict__ w_mv2s2, const float* __restrict__ w_s2,
             const float* __restrict__ b_s2,
             float* __restrict__ out_mv, float* __restrict__ out_s)
{
    __shared__ float hmv[64*16];
    __shared__ float hs[128];
    __shared__ float mv1[256], s1[64];
    __shared__ float nm[256], nsd[64];
    __shared__ float mm[32*16], ms[128];
    __shared__ float red[32];

    const int item = blockIdx.x;
    const int tid = threadIdx.x;
    const float* Ob = Oc + (size_t)item * NH * NF;

    for (int idx = tid; idx < NH*NF; idx += 128) {
        const int h = idx / NF, f = idx % NF;
        const float v = Ob[idx];
        if (f < 128) hmv[(h*8 + (f>>4))*16 + (f&15)] = v;
        else         hs[h*16 + (f - 128)] = v;
    }
    __syncthreads();

    // equi_linear(out) + residual
    for (int idx = tid; idx < 256; idx += 128) {
        const int o = idx >> 4, xg = idx & 15;
        const float* wr = w_mv_out + ((size_t)GRADE_c[xg]*16 + o)*64;
        float a = 0.f;
        for (int i = 0; i < 64; ++i) a += wr[i] * hmv[i*16 + xg];
        if (xg == 0) {
            const float* w2 = w_s2mv_out + (size_t)o * 128;
            for (int s = 0; s < 128; ++s) a += w2[s] * hs[s];
        }
        mv1[idx] = mv[(size_t)item*256 + idx] + a;
    }
    if (tid < 64) {
        const int o = tid;
        float a = b_s_out[o];
        const float* wr = w_s_out + (size_t)o * 128;
        for (int s = 0; s < 128; ++s) a += wr[s] * hs[s];
        const float* wm = w_mv2s_out + (size_t)o * 64;
        for (int i = 0; i < 64; ++i) a += wm[i] * hmv[i*16];
        s1[o] = sc[(size_t)item*64 + o] + a;
    }
    __syncthreads();

    // equi_norm
    if (tid < 16) {
        float a = 0.f;
        for (int j = 0; j < 16; ++j) { float v = mv1[tid*16 + j]; a += v*v*INNER_c[j]; }
        red[tid] = a;
    }
    __syncthreads();
    if (tid == 0) {
        float m = 0.f;
        for (int i = 0; i < 16; ++i) m += red[i];
        red[16] = 1.0f / sqrtf(m * (1.0f/16.0f) + 0.01f);
    }
    if (tid == 1) {
        float mu = 0.f;
        for (int i = 0; i < 64; ++i) mu += s1[i];
        mu *= (1.0f/64.0f);
        float v = 0.f;
        for (int i = 0; i < 64; ++i) { float d = s1[i] - mu; v += d*d; }
        v *= (1.0f/64.0f);
        red[17] = mu; red[18] = rsqrtf(v + 1e-5f);
    }
    __syncthreads();
    for (int idx = tid; idx < 256; idx += 128) nm[idx] = mv1[idx] * red[16];
    if (tid < 64) nsd[tid] = (s1[tid] - red[17]) * red[18];
    __syncthreads();

    // equi_linear(1): 32 mv channels, 128 scalar
    for (int idx = tid; idx < 512; idx += 128) {
        const int o = idx >> 4, xg = idx & 15;
        const float* wr = w_mv1 + ((size_t)GRADE_c[xg]*32 + o)*16;
        float a = 0.f;
        #pragma unroll
        for (int i = 0; i < 16; ++i) a += wr[i] * nm[i*16 + xg];
        if (xg == 0) {
            const float* w2 = w_s2mv1 + (size_t)o * 64;
            for (int s = 0; s < 64; ++s) a += w2[s] * nsd[s];
        }
        mm[idx] = a;
    }
    if (tid < 128) {
        const int o = tid;
        float a = b_s1[o];
        const float* wr = w_s1 + (size_t)o * 64;
        for (int s = 0; s < 64; ++s) a += wr[s] * nsd[s];
        const float* wm = w_mv2s1 + (size_t)o * 16;
        #pragma unroll
        for (int i = 0; i < 16; ++i) a += wm[i] * nm[i*16];
        ms[o] = a;
    }
    __syncthreads();

    // gating
    if (tid < 32) red[tid] = gelu_tanh(mm[tid*16]);
    __syncthreads();
    for (int idx = tid; idx < 512; idx += 128) mm[idx] *= red[idx >> 4];
    if (tid < 128) ms[tid] = gelu_tanh(ms[tid]);
    __syncthreads();

    // equi_linear(2) + residual out
    for (int idx = tid; idx < 256; idx += 128) {
        const int o = idx >> 4, xg = idx & 15;
        const float* wr = w_mv2 + ((size_t)GRADE_c[xg]*16 + o)*32;
        float a = 0.f;
        for (int i = 0; i < 32; ++i) a += wr[i] * mm[i*16 + xg];
        if (xg == 0) {
            const float* w2 = w_s2mv2 + (size_t)o * 128;
            for (int s = 0; s < 128; ++s) a += w2[s] * ms[s];
        }
        out_mv[(size_t)item*256 + idx] = mv1[idx] + a;
    }
    if (tid < 64) {
        const int o = tid;
        float a = b_s2[o];
        const float* wr = w_s2 + (size_t)o * 128;
        for (int s = 0; s < 128; ++s) a += wr[s] * ms[s];
        const float* wm = w_mv2s2 + (size_t)o * 32;
        for (int i = 0; i < 32; ++i) a += wm[i] * mm[i*16];
        out_s[(size_t)item*64 + o] = s1[o] + a;
    }
}

// ---------------------------------------------------------------------------
extern "C" void kernel_launch(void* const* d_in, const int* in_sizes, int n_in,
                              void* d_out, int out_size, void* d_ws, size_t ws_size,
                              hipStream_t stream) {
    const float* mv        = (const float*)d_in[0];
    const float* sc        = (const float*)d_in[1];
    const float* pz        = (const float*)d_in[2];
    const float* w_mv_qkv  = (const float*)d_in[3];
    const float* w_s2mv_qkv= (const float*)d_in[4];
    const float* w_mv2s_qkv= (const float*)d_in[5];
    const float* w_s_qkv   = (const float*)d_in[6];
    const float* b_s_qkv   = (const float*)d_in[7];
    const float* w_mv_out  = (const float*)d_in[8];
    const float* w_s2mv_out= (const float*)d_in[9];
    const float* w_mv2s_out= (const float*)d_in[10];
    const float* w_s_out   = (const float*)d_in[11];
    const float* b_s_out   = (const float*)d_in[12];
    const float* ln_z_g    = (const float*)d_in[13];
    const float* ln_z_b    = (const float*)d_in[14];
    const float* w_pb      = (const float*)d_in[15];
    const float* w_mv1     = (const float*)d_in[16];
    const float* w_s2mv1   = (const float*)d_in[17];
    const float* w_mv2s1   = (const float*)d_in[18];
    const float* w_s1      = (const float*)d_in[19];
    const float* b_s1      = (const float*)d_in[20];
    const float* w_mv2     = (const float*)d_in[21];
    const float* w_s2mv2   = (const float*)d_in[22];
    const float* w_mv2s2   = (const float*)d_in[23];
    const float* w_s2      = (const float*)d_in[24];
    const float* b_s2      = (const float*)d_in[25];

    float* ws = (float*)d_ws;
    const size_t szQ = (size_t)NB * NH * NF * NN;   // 1,179,648 floats each
    float* Qp   = ws;
    float* Kp   = Qp + szQ;
    float* Vc   = Kp + szQ;
    float* Oc   = Vc + szQ;
    float* bias = Oc + szQ;                          // NB*NH*NN*NN floats

    float* out_mv = (float*)d_out;
    float* out_s  = out_mv + (size_t)NB * NN * 256;

    ka_qkv<<<NB * NN, 384, 0, stream>>>(mv, sc, w_mv_qkv, w_s2mv_qkv, w_mv2s_qkv,
                                        w_s_qkv, b_s_qkv, Qp, Kp, Vc);
    kb_bias<<<(NB * NN * NN) / 8, 256, 0, stream>>>(pz, ln_z_g, ln_z_b, w_pb, bias);
    kc_attn<<<dim3(NN / 16, NH, NB), 32, 0, stream>>>(Qp, Kp, Vc, bias, Oc);
    kd_post<<<NB * NN, 128, 0, stream>>>(Oc, mv, sc,
                                         w_mv_out, w_s2mv_out, w_mv2s_out, w_s_out, b_s_out,
                                         w_mv1, w_s2mv1, w_mv2s1, w_s1, b_s1,
                                         w_mv2, w_s2mv2, w_mv2s2, w_s2, b_s2,
                                         out_mv, out_s);
}